// NonMaxSupression_72524817760380
// MI455X (gfx1250) — compile-verified
//
#include <hip/hip_runtime.h>

#define NBOX 6000
#define NOUT 2000
#define IOU_THR 0.8f
#define BDIM 1024

typedef unsigned int u32;
typedef unsigned long long u64;
typedef u32 u32x4 __attribute__((ext_vector_type(4)));
typedef u32 u32x8 __attribute__((ext_vector_type(8)));

// ---- CDNA5 Tensor Data Mover: 1D contiguous DWORD tile, global -> LDS ----
// D# per CDNA5 ISA. Issued once per wave (EXEC ignored),
// so callers must guard to a single wave. Tracked by TENSORcnt.
__device__ __forceinline__ void tdm_load_1d_dwords(u32 lds_byte_off, const void* gptr, u32 n_dwords) {
    const u64 ga = (u64)gptr;
    u32x4 g0;
    g0.x = 1u;                                              // count=1, is_restore=0, gather off
    g0.y = lds_byte_off;                                    // lds_addr
    g0.z = (u32)ga;                                         // global_addr[31:0]
    g0.w = (u32)((ga >> 32) & 0x01FFFFFFu) | (2u << 30);    // global_addr[56:32] | type=2
    u32x8 g1;
    g1.s0 = (2u << 16);                                     // wg_mask=0, data_size=2 (4B), no extras
    g1.s1 = (n_dwords & 0xFFFFu) << 16;                     // tensor_dim0[15:0]   (bits 63:48)
    g1.s2 = (n_dwords >> 16) | (1u << 16);                  // tensor_dim0[31:16], tensor_dim1=1
    g1.s3 = (n_dwords & 0xFFFFu) << 16;                     // tile_dim0           (bits 127:112)
    g1.s4 = 1u;                                             // tile_dim1=1, tile_dim2=0
    g1.s5 = n_dwords;                                       // tensor_dim0_stride[31:0]
    g1.s6 = 0u;                                             // stride0[47:32]=0, stride1[15:0]=0
    g1.s7 = 0u;                                             // stride1[47:16]=0
    asm volatile("tensor_load_to_lds %0, %1" :: "s"(g0), "s"(g1) : "memory");
}

// ---- CDNA5 async global->LDS gather (per-lane LDS dest = scatter-by-rank), ASYNCcnt-tracked ----
__device__ __forceinline__ void async_ld_b128(unsigned lds_byte_off, unsigned glb_byte_off,
                                              const void* sbase) {
    asm volatile("global_load_async_to_lds_b128 %0, %1, %2"
                 :: "v"(lds_byte_off), "v"(glb_byte_off), "s"(sbase)
                 : "memory");
}
__device__ __forceinline__ void wait_async0() {
    asm volatile("s_wait_asynccnt 0" ::: "memory");
}

__device__ __forceinline__ float iou_pair(float ay1, float ax1, float ay2, float ax2, float aA,
                                          float by1, float bx1, float by2, float bx2, float bA) {
    float ih = fminf(ay2, by2) - fmaxf(ay1, by1);
    float iw = fminf(ax2, bx2) - fmaxf(ax1, bx1);
    ih = fmaxf(ih, 0.0f);
    iw = fmaxf(iw, 0.0f);
    float inter = ih * iw;
    float uni = aA + bA - inter;
    return inter / fmaxf(uni, 1e-12f);
}

__global__ void nms_zero_out(float* __restrict__ out, int n) {
    int i = blockIdx.x * blockDim.x + threadIdx.x;
    if (i < n) out[i] = 0.0f;
}

__global__ __launch_bounds__(BDIM) void nms_batch_kernel(const float* __restrict__ boxes,
                                                         const float* __restrict__ scores,
                                                         float* __restrict__ out) {
    // Whole batch resident in LDS (CDNA5: 320KB/WGP; we use ~144KB).
    __shared__ float s_scores[NBOX];                    // 24 KB
    __shared__ __align__(16) float s_boxes[NBOX * 4];   // 96 KB, sorted by descending score
    __shared__ int s_keep[NBOX];                        // 24 KB
    __shared__ int s_wsum[BDIM / 32];                   // per-wave scan partials
    __shared__ int s_running;

    const int b = blockIdx.x;
    const int tid = threadIdx.x;
    const int lane = tid & 31;
    const int wid = tid >> 5;
    const float* gscores = scores + (size_t)b * NBOX;
    const float* gboxes = boxes + (size_t)b * NBOX * 4;

    // ---- Phase 0: TDM DMA: full score vector -> LDS in one descriptor (wave 0 issues) ----
    if (wid == 0) {
        tdm_load_1d_dwords((u32)(size_t)&s_scores[0], gscores, NBOX);
        __builtin_amdgcn_s_wait_tensorcnt(0);
    }
    __syncthreads();

    // ---- Phase 1: stable descending rank; async gather-scatter boxes into sorted LDS slots ----
    for (int i = tid; i < NBOX; i += BDIM) {
        float si = s_scores[i];
        int r = 0;
        for (int j = 0; j < NBOX; ++j) {
            float sj = s_scores[j];
            r += (sj > si) || ((sj == si) && (j < i));
        }
        // box i (16B in HBM) lands directly at sorted slot r in LDS via the async engine
        async_ld_b128((unsigned)(size_t)&s_boxes[4 * r], (unsigned)(i * 16), gboxes);
        s_keep[i] = 1;
    }
    wait_async0();
    __syncthreads();

    // ---- Phase 2: greedy suppression, tiled by wave32 width ----
    for (int i0 = 0; i0 < NBOX; i0 += 32) {
        const int tile_end = (i0 + 32 < NBOX) ? (i0 + 32) : NBOX;

        // 2a: wave 0 serially resolves suppression inside the tile (wave-sync, ballot-driven)
        if (wid == 0) {
            int i = i0 + lane;
            int ic = (i < NBOX) ? i : (NBOX - 1);
            int flag = (i < NBOX) ? s_keep[i] : 0;
            float my1 = s_boxes[4 * ic + 0];
            float mx1 = s_boxes[4 * ic + 1];
            float my2 = s_boxes[4 * ic + 2];
            float mx2 = s_boxes[4 * ic + 3];
            float mA = (my2 - my1) * (mx2 - mx1);
            for (int k = 0; k < 32; ++k) {
                unsigned live = (unsigned)__ballot(flag != 0);
                if (((live >> k) & 1u) && (lane > k)) {
                    int jk = i0 + k;
                    float ky1 = s_boxes[4 * jk + 0];
                    float kx1 = s_boxes[4 * jk + 1];
                    float ky2 = s_boxes[4 * jk + 2];
                    float kx2 = s_boxes[4 * jk + 3];
                    float kA = (ky2 - ky1) * (kx2 - kx1);
                    float v = iou_pair(my1, mx1, my2, mx2, mA, ky1, kx1, ky2, kx2, kA);
                    if (flag && (v > IOU_THR)) flag = 0;
                }
            }
            if (i < NBOX) s_keep[i] = flag;
        }
        __syncthreads();

        // 2b: all 32 waves sweep j > tile against the tile's survivors
        for (int j = tile_end + tid; j < NBOX; j += BDIM) {
            if (!s_keep[j]) continue;
            float jy1 = s_boxes[4 * j + 0];
            float jx1 = s_boxes[4 * j + 1];
            float jy2 = s_boxes[4 * j + 2];
            float jx2 = s_boxes[4 * j + 3];
            float jA = (jy2 - jy1) * (jx2 - jx1);
            int kill = 0;
            for (int m = i0; m < tile_end; ++m) {
                if (!s_keep[m]) continue;
                float ky1 = s_boxes[4 * m + 0];
                float kx1 = s_boxes[4 * m + 1];
                float ky2 = s_boxes[4 * m + 2];
                float kx2 = s_boxes[4 * m + 3];
                float kA = (ky2 - ky1) * (kx2 - kx1);
                float v = iou_pair(jy1, jx1, jy2, jx2, jA, ky1, kx1, ky2, kx2, kA);
                if (v > IOU_THR) { kill = 1; break; }
            }
            if (kill) s_keep[j] = 0;
        }
        __syncthreads();
    }

    // ---- Phase 3: compaction via wave32 ballot/popc scan, 2000-row cap ----
    if (tid == 0) s_running = 0;
    __syncthreads();
    float* gout = out + (size_t)b * NOUT * 4;
    for (int base = 0; base < NBOX; base += BDIM) {
        int i = base + tid;
        int k = (i < NBOX) ? s_keep[i] : 0;
        unsigned mask = (unsigned)__ballot(k != 0);
        int lexcl = __popc(mask & ((1u << lane) - 1u));
        if (lane == 0) s_wsum[wid] = __popc(mask);
        __syncthreads();
        if (wid == 0) {  // wave 0 scans the 32 per-wave totals (inclusive)
            int v = s_wsum[lane];
            for (int off = 1; off < 32; off <<= 1) {
                int t = __shfl_up(v, off);
                if (lane >= off) v += t;
            }
            s_wsum[lane] = v;
        }
        __syncthreads();
        int woff = (wid == 0) ? 0 : s_wsum[wid - 1];
        int pos = s_running + woff + lexcl;
        if (k && pos < NOUT) {
            const float4 bb = *(const float4*)&s_boxes[4 * i];
            *(float4*)&gout[4 * pos] = bb;
        }
        __syncthreads();
        if (tid == 0) s_running += s_wsum[(BDIM / 32) - 1];
        __syncthreads();
    }
}

extern "C" void kernel_launch(void* const* d_in, const int* in_sizes, int n_in,
                              void* d_out, int out_size, void* d_ws, size_t ws_size,
                              hipStream_t stream) {
    const float* boxes = (const float*)d_in[0];   // (B, N, 4) f32
    const float* scores = (const float*)d_in[1];  // (B, N)    f32
    float* out = (float*)d_out;                   // (B, 2000, 4) f32
    (void)d_ws; (void)ws_size; (void)n_in;

    const int B = in_sizes[1] / NBOX;

    nms_zero_out<<<(out_size + 255) / 256, 256, 0, stream>>>(out, out_size);
    nms_batch_kernel<<<B, BDIM, 0, stream>>>(boxes, scores, out);
}